// AttenSurfaceClassifier_66855460929791
// MI455X (gfx1250) — compile-verified
//
#include <hip/hip_runtime.h>

// ---------------------------------------------------------------------------
// Fully fused AttenSurfaceClassifier for gfx1250 (MI455X).
// One workgroup processes 8 points end-to-end using fp32 WMMA (16x16x4).
// Activations are column-major [col][C+pad] in LDS so WMMA B fragments and
// D stores are contiguous b64 LDS ops.
// ---------------------------------------------------------------------------

typedef float v2f __attribute__((ext_vector_type(2)));
typedef float v8f __attribute__((ext_vector_type(8)));

#define NVIEW 6
#define PTS 8                 // points per workgroup
#define COLS (NVIEW * PTS)    // 48 (v,p) columns; col = v*8 + p
#define NTHREADS 256
#define NWAVES 8

// padded channel strides (even -> 8B alignment for float2 LDS ops)
#define XPAD 258    // 256 + 2
#define Y0PAD 1026  // 1024 + 2
#define MPAD 514    // 512 + 2
#define Y2PAD 258   // 256 + 2
#define Y3PAD 130   // 128 + 2

// LDS layout (floats)
#define XOFF 0                         // X: 48 x 258 = 12384
#define Y0OFF 12384                    // Y0: 48 x 1026 = 49248
#define ATTOFF (12384 + 49248)         // att scratch: 8 x 40
#define LDS_FLOATS (ATTOFF + 352)
// aliases (regions dead by the time these are written)
#define YSUMOFF XOFF                   // 16 x 512 partial view-sums (X region)
#define YMEANOFF Y0OFF                 // 16 x 514, cols(v,p) 8..15 zeroed
#define Y2OFF (Y0OFF + 16 * MPAD)      // 16 x 258
#define Y3OFF (Y2OFF + 16 * Y2PAD)     // 16 x 130

__device__ __forceinline__ float leakyf(float x) { return x >= 0.f ? x : 0.01f * x; }

// ---------------------------------------------------------------------------
// D[Cout x 16*NCOLT] = W[Cout x Cin] * X[16*NCOLT cols x Cin (col-major)]
// fp32 WMMA 16x16x4: lane m=lane&15, hi=lane>>4, kq=hi*2
//   A frag: a[i] = W[mt*16+m][k+kq+i]          (global, contiguous pair)
//   B frag: b[i] = X[ncol][k+kq+i]             (LDS, contiguous pair)
//   D frag: vgpr j -> row mt*16+j+hi*8, col ncol
// ---------------------------------------------------------------------------
template <int COUT, int CIN, int CINPAD, int COUTPAD, int NCOLT, bool LEAKY, int VALIDOUT>
__device__ __forceinline__ void conv_gemm(const float* __restrict__ W,
                                          const float* __restrict__ bias,
                                          const float* Xl, float* Yl,
                                          int wave, int lane)
{
  const int m  = lane & 15;
  const int hi = lane >> 4;   // 0 or 1
  const int kq = hi * 2;      // 0 or 2
  constexpr int NT = (COUT / 16) * NCOLT;
  for (int t = wave; t < NT; t += NWAVES) {
    const int mt = t / NCOLT;
    const int nt = t - mt * NCOLT;
    const int ncol = nt * 16 + m;
    v8f acc;
#pragma unroll
    for (int j = 0; j < 8; ++j) acc[j] = bias[mt * 16 + j + hi * 8];
    const float* wrow = W + (size_t)(mt * 16 + m) * CIN + kq;
    const float* xcol = Xl + ncol * CINPAD + kq;
    // prefetch next tile's weight row into cache while this tile computes
    const int tn = t + NWAVES;
    if (tn < NT) __builtin_prefetch(W + (size_t)((tn / NCOLT) * 16 + m) * CIN, 0, 1);
#pragma unroll 8
    for (int k = 0; k < CIN; k += 4) {
      const float2 av = *(const float2*)(wrow + k);   // global b64 (L2-resident)
      const float2 bv = *(const float2*)(xcol + k);   // ds b64, contiguous
      v2f a, b;
      a.x = av.x; a.y = av.y;
      b.x = bv.x; b.y = bv.y;
      acc = __builtin_amdgcn_wmma_f32_16x16x4_f32(false, a, false, b,
                                                  (short)0, acc, false, false);
    }
    float* ycol = Yl + ncol * COUTPAD + mt * 16 + hi * 8;
#pragma unroll
    for (int j = 0; j < 8; j += 2) {                  // 4x ds_store_b64
      float v0 = acc[j], v1 = acc[j + 1];
      if (LEAKY) { v0 = leakyf(v0); v1 = leakyf(v1); }
      if (VALIDOUT < 16 * NCOLT && ncol >= VALIDOUT) { v0 = 0.f; v1 = 0.f; }
      *(float2*)(ycol + j) = make_float2(v0, v1);
    }
  }
}

// ---------------------------------------------------------------------------
// In-place per-point 6x6 cross-view attention on F[col][CPAD], col = v*8+p.
// ---------------------------------------------------------------------------
template <int C, int CPAD>
__device__ __forceinline__ void view_attn_lds(float* F, float* attS, int tid)
{
  const int wave = tid >> 5, lane = tid & 31;
  // Gram: wave w owns point p=w; 36 dot products of length C (contiguous b64)
  {
    const int p = wave;
    for (int q = 0; q < 36; ++q) {
      const int v = q / 6, w = q - v * 6;
      const float2* fv = (const float2*)(F + (v * PTS + p) * CPAD);
      const float2* fw = (const float2*)(F + (w * PTS + p) * CPAD);
      float s = 0.f;
#pragma unroll 4
      for (int c2 = lane; c2 < C / 2; c2 += 32) {
        const float2 x = fv[c2], y = fw[c2];
        s += x.x * y.x + x.y * y.y;
      }
#pragma unroll
      for (int off = 16; off > 0; off >>= 1)
        s += __shfl_down(s, off, 32);
      if (lane == 0) attS[p * 40 + q] = s;
    }
  }
  __syncthreads();
  // softmax over w for each of the 48 (p,v) rows
  if (tid < PTS * NVIEW) {
    const int p = tid / NVIEW, v = tid - p * NVIEW;
    float* r = attS + p * 40 + v * 6;
    float mx = r[0];
#pragma unroll
    for (int w = 1; w < 6; ++w) mx = fmaxf(mx, r[w]);
    float e[6], sum = 0.f;
#pragma unroll
    for (int w = 0; w < 6; ++w) { e[w] = __expf(r[w] - mx); sum += e[w]; }
    const float inv = 1.f / sum;
#pragma unroll
    for (int w = 0; w < 6; ++w) r[w] = e[w] * inv;
  }
  __syncthreads();
  // apply: out[v,c] = sum_w att[v,w] * f[w,c]
  // grid stride is a multiple of 8 -> each thread's point p is fixed: hoist att
  {
    const int p = tid & 7;
    float att[36];
#pragma unroll
    for (int q = 0; q < 36; ++q) att[q] = attS[p * 40 + q];
    for (int idx = tid; idx < PTS * (C / 2); idx += NTHREADS) {
      const int c2 = idx >> 3;   // float2 channel index
      float2 f[6];
#pragma unroll
      for (int w = 0; w < 6; ++w)
        f[w] = *(const float2*)(F + (w * PTS + p) * CPAD + 2 * c2);
#pragma unroll
      for (int v = 0; v < 6; ++v) {
        float sx = 0.f, sy = 0.f;
#pragma unroll
        for (int w = 0; w < 6; ++w) {
          sx += att[v * 6 + w] * f[w].x;
          sy += att[v * 6 + w] * f[w].y;
        }
        *(float2*)(F + (v * PTS + p) * CPAD + 2 * c2) = make_float2(sx, sy);
      }
    }
  }
  __syncthreads();
}

// ---------------------------------------------------------------------------
__global__ void __launch_bounds__(NTHREADS, 1)
atten_surface_fused(const float* __restrict__ feature,
                    const float* __restrict__ W0, const float* __restrict__ b0,
                    const float* __restrict__ W1, const float* __restrict__ b1,
                    const float* __restrict__ W2, const float* __restrict__ b2,
                    const float* __restrict__ W3, const float* __restrict__ b3,
                    const float* __restrict__ W4, const float* __restrict__ b4,
                    float* __restrict__ out, int n)
{
  extern __shared__ float lds[];
  float* Xl    = lds + XOFF;
  float* Y0    = lds + Y0OFF;
  float* attS  = lds + ATTOFF;
  float* Ysum  = lds + YSUMOFF;
  float* Ymean = lds + YMEANOFF;
  float* Y2    = lds + Y2OFF;
  float* Y3    = lds + Y3OFF;

  const int tid = threadIdx.x;
  const int wave = tid >> 5, lane = tid & 31;
  const int n0 = blockIdx.x * PTS;

  // ---- load feature tile (6 x 256 x 8), float4 over n, scatter to col-major ----
#pragma unroll
  for (int it = 0; it < 12; ++it) {
    const int u = tid + it * NTHREADS;      // quad id, 0..3071
    const int v = u / 512;                  // 2 quads per (v,c)
    const int rem = u - v * 512;
    const int c = rem >> 1;
    const int q = rem & 1;
    const float4 val = *(const float4*)(feature + (size_t)(v * 256 + c) * n + n0 + q * 4);
    float* base = Xl + c + (v * PTS + q * 4) * XPAD;
    base[0 * XPAD] = val.x;
    base[1 * XPAD] = val.y;
    base[2 * XPAD] = val.z;
    base[3 * XPAD] = val.w;
  }
  __syncthreads();

  // ---- attention on input (c=256) ----
  view_attn_lds<256, XPAD>(Xl, attS, tid);

  // ---- conv0 (1024x256) + leaky ----
  conv_gemm<1024, 256, XPAD, Y0PAD, 3, true, COLS>(W0, b0, Xl, Y0, wave, lane);
  __syncthreads();

  // ---- attention on Y0 (c=1024) ----
  view_attn_lds<1024, Y0PAD>(Y0, attS, tid);

  // ---- conv1 (512x1024) + leaky + partial view-sum (3 col tiles in regs) ----
  {
    const int m = lane & 15;
    const int hi = lane >> 4;
    const int kq = hi * 2;
    for (int mt = wave; mt < 32; mt += NWAVES) {
      float ssum[8];
#pragma unroll
      for (int j = 0; j < 8; ++j) ssum[j] = 0.f;
      const float* wrow = W1 + (size_t)(mt * 16 + m) * 1024 + kq;
      const int mtn = mt + NWAVES;
      if (mtn < 32) __builtin_prefetch(W1 + (size_t)(mtn * 16 + m) * 1024, 0, 1);
#pragma unroll
      for (int nt = 0; nt < 3; ++nt) {
        const int ncol = nt * 16 + m;
        v8f acc;
#pragma unroll
        for (int j = 0; j < 8; ++j) acc[j] = b1[mt * 16 + j + hi * 8];
        const float* xcol = Y0 + ncol * Y0PAD + kq;
#pragma unroll 8
        for (int k = 0; k < 1024; k += 4) {
          const float2 av = *(const float2*)(wrow + k);
          const float2 bv = *(const float2*)(xcol + k);
          v2f a, b;
          a.x = av.x; a.y = av.y;
          b.x = bv.x; b.y = bv.y;
          acc = __builtin_amdgcn_wmma_f32_16x16x4_f32(false, a, false, b,
                                                      (short)0, acc, false, false);
        }
#pragma unroll
        for (int j = 0; j < 8; ++j) ssum[j] += leakyf(acc[j]);
      }
      float* ycol = Ysum + m * 512 + mt * 16 + hi * 8;
#pragma unroll
      for (int j = 0; j < 8; j += 2)
        *(float2*)(ycol + j) = make_float2(ssum[j], ssum[j + 1]);
    }
  }
  __syncthreads();

  // ---- finalize mean over views: Ymean[16 cols][514], cols 8..15 zeroed ----
  for (int idx = tid; idx < 16 * 512; idx += NTHREADS) {
    const int nc = idx >> 9;      // col 0..15
    const int o  = idx & 511;     // channel
    float v = 0.f;
    if (nc < 8)
      v = (Ysum[nc * 512 + o] + Ysum[(nc + 8) * 512 + o]) * (1.f / 6.f);
    Ymean[nc * MPAD + o] = v;
  }
  __syncthreads();

  // ---- conv2 (256x512) + leaky ----
  conv_gemm<256, 512, MPAD, Y2PAD, 1, true, PTS>(W2, b2, Ymean, Y2, wave, lane);
  __syncthreads();

  // ---- conv3 (128x256) + leaky ----
  conv_gemm<128, 256, Y2PAD, Y3PAD, 1, true, PTS>(W3, b3, Y2, Y3, wave, lane);
  __syncthreads();

  // ---- conv4 (1x128), final output ----
  if (tid < PTS) {
    float s = b4[0];
    const float2* y = (const float2*)(Y3 + tid * Y3PAD);
#pragma unroll 8
    for (int c2 = 0; c2 < 64; ++c2) {
      const float2 wv = *(const float2*)(W4 + 2 * c2);
      const float2 yv = y[c2];
      s += wv.x * yv.x + wv.y * yv.y;
    }
    out[n0 + tid] = s;
  }
}

// ---------------------------------------------------------------------------
extern "C" void kernel_launch(void* const* d_in, const int* in_sizes, int n_in,
                              void* d_out, int out_size, void* d_ws, size_t ws_size,
                              hipStream_t stream) {
  const float* feature = (const float*)d_in[0];
  const float* W0 = (const float*)d_in[1];
  const float* b0 = (const float*)d_in[2];
  const float* W1 = (const float*)d_in[3];
  const float* b1 = (const float*)d_in[4];
  const float* W2 = (const float*)d_in[5];
  const float* b2 = (const float*)d_in[6];
  const float* W3 = (const float*)d_in[7];
  const float* b3 = (const float*)d_in[8];
  const float* W4 = (const float*)d_in[9];
  const float* b4 = (const float*)d_in[10];
  float* out = (float*)d_out;

  const int n = in_sizes[0] / (NVIEW * 256);   // 16384
  const int nblocks = n / PTS;                 // 2048
  const size_t shmem = (size_t)LDS_FLOATS * sizeof(float);  // ~248 KB of 320 KB LDS

  (void)hipFuncSetAttribute((const void*)atten_surface_fused,
                            hipFuncAttributeMaxDynamicSharedMemorySize, (int)shmem);
  atten_surface_fused<<<nblocks, NTHREADS, shmem, stream>>>(
      feature, W0, b0, W1, b1, W2, b2, W3, b3, W4, b4, out, n);
}